// CorrectedRSEBERTAttention_41446434406929
// MI455X (gfx1250) — compile-verified
//
#include <hip/hip_runtime.h>
#include <math.h>

typedef float v8f __attribute__((ext_vector_type(8)));
typedef float v4f __attribute__((ext_vector_type(4)));
typedef float v2f __attribute__((ext_vector_type(2)));

#define B_      4
#define H_      12
#define S_      1024
#define D_      64
#define HALF_   32
#define LAMBDA_ 0.01f
#define EPS_    1e-6f
#define CLAMP_  20.0f
#define QTILES  (S_ / 16)          // 64 query tiles per (b,h)
#define WPB     4                  // waves per block
#define LSTR    20                 // padded LDS row stride: 16B-aligned rows,
                                   // 20*dm % 64 != 0 for dm in [1,15] -> conflict-free

__global__ __launch_bounds__(32 * WPB)
void rse_stickbreak_attn(const float* __restrict__ q,
                         const float* __restrict__ k,
                         const float* __restrict__ v,
                         const float* __restrict__ cosc,
                         const float* __restrict__ sinc,
                         float* __restrict__ out)
{
    __shared__ __align__(16) float lds[WPB][16 * LSTR + 16];

    const int wave = threadIdx.x >> 5;
    const int lane = threadIdx.x & 31;
    const int tile = blockIdx.x * WPB + wave;     // 0 .. B*H*QTILES-1
    const int qt   = tile % QTILES;
    const int bh   = tile / QTILES;

    float* wlds = &lds[wave][0];                  // 16x16 beta/weight tile
    float* dlds = &lds[wave][16 * LSTR];          // 16 row denominators

    const int hi = lane >> 4;                     // 0 or 1 (lane half)
    const int ln = lane & 15;

    const float* qb = q + (size_t)bh * S_ * D_;
    const float* kb = k + (size_t)bh * S_ * D_;
    const float* vb = v + (size_t)bh * S_ * D_;
    float*       ob = out + (size_t)bh * S_ * D_;

    // ---- A fragments: RoPE'd q tile, held for the whole k loop.
    // A 16x4 f32 layout: M = lane&15; VGPR j holds K = 2*(lane>>4)+j.
    // Chunk c covers head-dim K-range [4c, 4c+4); (d, d+1), d even, is a RoPE pair
    // and is 8B-aligned-contiguous -> one b64 load.
    const int qrow = qt * 16 + ln;
    v2f afrag[16];
#pragma unroll
    for (int c = 0; c < 16; ++c) {
        const int d  = 4 * c + 2 * hi;
        const v2f qp = *(const v2f*)(qb + qrow * D_ + d);           // (even, odd)
        const float cc = cosc[qrow * HALF_ + (d >> 1)];
        const float ss = sinc[qrow * HALF_ + (d >> 1)];
        afrag[c].x = qp.x * cc - qp.y * ss;
        afrag[c].y = qp.y * cc + qp.x * ss;
    }

    v8f acc0 = (v8f)0.0f, acc1 = (v8f)0.0f, acc2 = (v8f)0.0f, acc3 = (v8f)0.0f;
    float rem   = 1.0f;        // remaining mass (valid in lanes 0..15; row = lane)
    float denom = 0.0f;
    const float scale = 0.125f;   // 1/sqrt(64)

    for (int kt = 0; kt <= qt; ++kt) {
        if (kt < qt) {   // near-scope prefetch of next k/v tiles (global_prefetch_b8)
            __builtin_prefetch(&kb[(kt + 1) * 16 * D_ + lane * 32], 0, 3);
            __builtin_prefetch(&vb[(kt + 1) * 16 * D_ + lane * 32], 0, 3);
        }

        // ---- logits tile: Q_r (16xD) x K_r^T (Dx16) via 16 chained f32 WMMAs.
        // B 4x16 f32 layout: N = lane&15; VGPR j holds K = 2*(lane>>4)+j.
        const int krow = kt * 16 + ln;
        v8f cacc = (v8f)0.0f;
#pragma unroll
        for (int c = 0; c < 16; ++c) {
            const int d  = 4 * c + 2 * hi;
            const v2f kp = *(const v2f*)(kb + krow * D_ + d);       // b64 load
            const float cc = cosc[krow * HALF_ + (d >> 1)];
            const float ss = sinc[krow * HALF_ + (d >> 1)];
            v2f bfrag;
            bfrag.x = kp.x * cc - kp.y * ss;
            bfrag.y = kp.y * cc + kp.x * ss;
            cacc = __builtin_amdgcn_wmma_f32_16x16x4_f32(
                false, afrag[c], false, bfrag, (short)0, cacc, false, false);
        }

        // ---- beta = sigmoid(clip(logit - lambda*|dpos|)), causal mask; -> LDS.
        // C layout: VGPR r holds (M = r + 8*hi, N = lane&15).
#pragma unroll
        for (int r = 0; r < 8; ++r) {
            const int m    = r + 8 * hi;
            const int qpos = qt * 16 + m;
            const int kpos = kt * 16 + ln;
            float x = cacc[r] * scale - LAMBDA_ * fabsf((float)(qpos - kpos));
            const bool fut = kpos > qpos;
            x = fminf(fmaxf(x, -CLAMP_), CLAMP_);
            float beta = 1.0f / (1.0f + __expf(-x));
            if (fut) beta = 0.0f;
            wlds[m * LSTR + ln] = beta;
        }
        asm volatile("s_wait_dscnt 0" ::: "memory");
        __builtin_amdgcn_wave_barrier();

        // ---- sequential stick-breaking scan along k (one lane per query row).
        // Pull the whole row with 4x ds_load_b128, recur in registers, write back.
        if (lane < 16) {
            v4f* rowp = (v4f*)(wlds + lane * LSTR);
            v4f w0 = rowp[0], w1 = rowp[1], w2 = rowp[2], w3 = rowp[3];
            float r_   = rem;
            float dsum = 0.0f;
#define SB_STEP(VEC, I)                                   \
            { const float w = VEC[I] * r_;                \
              r_ = fmaxf(r_ * (1.0f - w), EPS_);          \
              VEC[I] = w;  dsum += w; }
            SB_STEP(w0, 0) SB_STEP(w0, 1) SB_STEP(w0, 2) SB_STEP(w0, 3)
            SB_STEP(w1, 0) SB_STEP(w1, 1) SB_STEP(w1, 2) SB_STEP(w1, 3)
            SB_STEP(w2, 0) SB_STEP(w2, 1) SB_STEP(w2, 2) SB_STEP(w2, 3)
            SB_STEP(w3, 0) SB_STEP(w3, 1) SB_STEP(w3, 2) SB_STEP(w3, 3)
#undef SB_STEP
            rowp[0] = w0; rowp[1] = w1; rowp[2] = w2; rowp[3] = w3;
            rem    = r_;
            denom += dsum;
        }
        asm volatile("s_wait_dscnt 0" ::: "memory");
        __builtin_amdgcn_wave_barrier();

        // ---- out(16x64) += W(16x16) @ V(16x64): 4 d-chunks x 4 K-chunks of WMMA.
        v2f awf[4];
#pragma unroll
        for (int kc = 0; kc < 4; ++kc)   // contiguous pair -> ds_load_b64
            awf[kc] = *(const v2f*)(wlds + ln * LSTR + 4 * kc + 2 * hi);
#pragma unroll
        for (int kc = 0; kc < 4; ++kc) {
            const int vr = kt * 16 + 4 * kc + 2 * hi;   // B VGPR0 row; VGPR1 is vr+1
            v2f b0, b1, b2, b3;
            b0.x = vb[vr * D_ +  0 + ln];  b0.y = vb[(vr + 1) * D_ +  0 + ln];
            b1.x = vb[vr * D_ + 16 + ln];  b1.y = vb[(vr + 1) * D_ + 16 + ln];
            b2.x = vb[vr * D_ + 32 + ln];  b2.y = vb[(vr + 1) * D_ + 32 + ln];
            b3.x = vb[vr * D_ + 48 + ln];  b3.y = vb[(vr + 1) * D_ + 48 + ln];
            acc0 = __builtin_amdgcn_wmma_f32_16x16x4_f32(
                false, awf[kc], false, b0, (short)0, acc0, false, false);
            acc1 = __builtin_amdgcn_wmma_f32_16x16x4_f32(
                false, awf[kc], false, b1, (short)0, acc1, false, false);
            acc2 = __builtin_amdgcn_wmma_f32_16x16x4_f32(
                false, awf[kc], false, b2, (short)0, acc2, false, false);
            acc3 = __builtin_amdgcn_wmma_f32_16x16x4_f32(
                false, awf[kc], false, b3, (short)0, acc3, false, false);
        }
    }

    // ---- epilogue: out /= max(rowsum(W), eps)
    if (lane < 16) dlds[lane] = fmaxf(denom, EPS_);
    asm volatile("s_wait_dscnt 0" ::: "memory");
    __builtin_amdgcn_wave_barrier();

#pragma unroll
    for (int r = 0; r < 8; ++r) {
        const int m  = r + 8 * hi;
        const float dn = dlds[m];
        const int orow = qt * 16 + m;
        ob[orow * D_ +  0 + ln] = acc0[r] / dn;
        ob[orow * D_ + 16 + ln] = acc1[r] / dn;
        ob[orow * D_ + 32 + ln] = acc2[r] / dn;
        ob[orow * D_ + 48 + ln] = acc3[r] / dn;
    }
}

extern "C" void kernel_launch(void* const* d_in, const int* in_sizes, int n_in,
                              void* d_out, int out_size, void* d_ws, size_t ws_size,
                              hipStream_t stream) {
    (void)in_sizes; (void)n_in; (void)out_size; (void)d_ws; (void)ws_size;
    const float* q    = (const float*)d_in[0];
    const float* k    = (const float*)d_in[1];
    const float* v    = (const float*)d_in[2];
    const float* cosc = (const float*)d_in[3];
    const float* sinc = (const float*)d_in[4];
    float* out = (float*)d_out;

    const int tiles  = B_ * H_ * QTILES;          // 3072 q-tiles, 1 wave each
    const int blocks = tiles / WPB;               // 768 blocks of 128 threads
    rse_stickbreak_attn<<<blocks, 32 * WPB, 0, stream>>>(q, k, v, cosc, sinc, out);
}